// CpoLoss_slow_11553462026767
// MI455X (gfx1250) — compile-verified
//
#include <hip/hip_runtime.h>
#include <hip/hip_bf16.h>
#include <math.h>

#define BLOCK 256
#define TILE_FLOATS 1024      // BLOCK lanes * 4 floats (b128) each = 4 KB tile
#define PIPE 4                // async tiles in flight
#define K_TOP 5

// ---- gfx1250 async global->LDS helpers (inline asm; portable across toolchains) ----
__device__ __forceinline__ unsigned lds_off_of(const void* p) {
  // low 32 bits of a flat pointer into LDS == LDS byte offset (aperture is in the high bits)
  return (unsigned)(unsigned long long)p;
}

__device__ __forceinline__ void async_ld_b128_nt(unsigned lds_byte_addr, int gbyte_off,
                                                 const float* sbase) {
  // GVS form: saddr(64b SGPR pair) + vaddr(32b VGPR offset); vdst holds LDS byte address.
  // Streamed-once data -> non-temporal hint keeps L2 clean for other rows.
  asm volatile("global_load_async_to_lds_b128 %0, %1, %2 th:TH_LOAD_NT"
               :: "v"(lds_byte_addr), "v"(gbyte_off), "s"(sbase)
               : "memory");
}

#define S_WAIT_ASYNCCNT_0() asm volatile("s_wait_asynccnt 0" ::: "memory")
#define S_WAIT_ASYNCCNT_1() asm volatile("s_wait_asynccnt 1" ::: "memory")
#define S_WAIT_ASYNCCNT_2() asm volatile("s_wait_asynccnt 2" ::: "memory")
#define S_WAIT_ASYNCCNT_3() asm volatile("s_wait_asynccnt 3" ::: "memory")
#define S_WAIT_DSCNT_0()    asm volatile("s_wait_dscnt 0"    ::: "memory")

__device__ __forceinline__ void wait_async_le(int n) {
  switch (n) {
    case 0:  S_WAIT_ASYNCCNT_0(); break;
    case 1:  S_WAIT_ASYNCCNT_1(); break;
    case 2:  S_WAIT_ASYNCCNT_2(); break;
    default: S_WAIT_ASYNCCNT_3(); break;
  }
}

// ---- sorted-descending top-5 insertion (register arrays, fully unrolled) ----
__device__ __forceinline__ void top5_insert(float v, int gi, float (&tv)[K_TOP], int (&ti)[K_TOP]) {
  if (v <= tv[K_TOP - 1]) return;
  tv[K_TOP - 1] = v; ti[K_TOP - 1] = gi;
#pragma unroll
  for (int k = K_TOP - 1; k > 0; --k) {
    if (tv[k] > tv[k - 1]) {
      float fv = tv[k]; tv[k] = tv[k - 1]; tv[k - 1] = fv;
      int   fi = ti[k]; ti[k] = ti[k - 1]; ti[k - 1] = fi;
    }
  }
}

__global__ __launch_bounds__(BLOCK)
void cpo_row_kernel(const float* __restrict__ logits,
                    const int*   __restrict__ target,
                    const float* __restrict__ mask,
                    float*       __restrict__ row_out,   // [rows] = row_loss * mask
                    int V)
{
  __shared__ float stage[PIPE][TILE_FLOATS];       // 16 KB async staging ring
  __shared__ float red_m[BLOCK], red_s[BLOCK];     // online-softmax reduce
  __shared__ float tmv[BLOCK * K_TOP];             // top-5 merge values
  __shared__ int   tmi[BLOCK * K_TOP];             // top-5 merge indices
  __shared__ float sTgt;

  const int row = blockIdx.x;
  const int tid = threadIdx.x;
  const float* rowp = logits + (size_t)row * (size_t)V;
  const int tgt = target[row];

  if (tid == 0) sTgt = 0.0f;

  // per-lane streaming state
  float m = -INFINITY, s = 0.0f;
  float tv[K_TOP]; int ti[K_TOP];
#pragma unroll
  for (int k = 0; k < K_TOP; ++k) { tv[k] = -INFINITY; ti[k] = -1; }
  float tgtv = -INFINITY;

  const int nFull = V / TILE_FLOATS;               // 49 full tiles for V=50257
  unsigned ldsBase[PIPE];
#pragma unroll
  for (int p = 0; p < PIPE; ++p) ldsBase[p] = lds_off_of(&stage[p][tid * 4]);
  const int laneByte = tid * 16;

  int issued = 0;
  // initial pipeline fill
  for (; issued < nFull && issued < PIPE; ++issued)
    async_ld_b128_nt(ldsBase[issued], issued * (TILE_FLOATS * 4) + laneByte, rowp);

  for (int t = 0; t < nFull; ++t) {
    wait_async_le(issued - t - 1);                 // tile t has landed in LDS
    const float4 vv = *reinterpret_cast<const float4*>(&stage[t % PIPE][tid * 4]);
    S_WAIT_DSCNT_0();                              // LDS read drained -> safe to recycle buffer
    if (issued < nFull) {                          // top up pipeline into the freed slot
      async_ld_b128_nt(ldsBase[issued % PIPE], issued * (TILE_FLOATS * 4) + laneByte, rowp);
      ++issued;
    }
    const int gbase = t * TILE_FLOATS + tid * 4;
    const float v4[4] = { vv.x, vv.y, vv.z, vv.w };
#pragma unroll
    for (int j = 0; j < 4; ++j) {
      const float v = v4[j];
      const int gi = gbase + j;
      if (v > m) { s = s * __expf(m - v) + 1.0f; m = v; }
      else       { s += __expf(v - m); }
      top5_insert(v, gi, tv, ti);
      if (gi == tgt) tgtv = v;
    }
  }
  // tail (V % 1024 = 81 elements): guarded non-temporal global loads
  for (int i = nFull * TILE_FLOATS + tid; i < V; i += BLOCK) {
    const float v = __builtin_nontemporal_load(&rowp[i]);
    if (v > m) { s = s * __expf(m - v) + 1.0f; m = v; }
    else       { s += __expf(v - m); }
    top5_insert(v, i, tv, ti);
    if (i == tgt) tgtv = v;
  }
  if (tgtv != -INFINITY) sTgt = tgtv;              // exactly one lane writes

  // ---- block reductions ----
  red_m[tid] = m; red_s[tid] = s;
#pragma unroll
  for (int k = 0; k < K_TOP; ++k) { tmv[tid * K_TOP + k] = tv[k]; tmi[tid * K_TOP + k] = ti[k]; }
  __syncthreads();

  for (int off = BLOCK / 2; off > 0; off >>= 1) {
    if (tid < off) {
      // online-softmax combine
      const float m1 = red_m[tid],       s1 = red_s[tid];
      const float m2 = red_m[tid + off], s2 = red_s[tid + off];
      const float mm = fmaxf(m1, m2);
      red_m[tid] = mm;
      red_s[tid] = s1 * __expf(m1 - mm) + s2 * __expf(m2 - mm);
      // merge two sorted top-5 lists (two-pointer)
      float a[K_TOP], b[K_TOP], ov[K_TOP];
      int   ai[K_TOP], bi[K_TOP], oi[K_TOP];
#pragma unroll
      for (int k = 0; k < K_TOP; ++k) {
        a[k] = tmv[tid * K_TOP + k];         ai[k] = tmi[tid * K_TOP + k];
        b[k] = tmv[(tid + off) * K_TOP + k]; bi[k] = tmi[(tid + off) * K_TOP + k];
      }
      int ia = 0, ib = 0;
#pragma unroll
      for (int k = 0; k < K_TOP; ++k) {
        if (a[ia] >= b[ib]) { ov[k] = a[ia]; oi[k] = ai[ia]; ++ia; }
        else                { ov[k] = b[ib]; oi[k] = bi[ib]; ++ib; }
      }
#pragma unroll
      for (int k = 0; k < K_TOP; ++k) { tmv[tid * K_TOP + k] = ov[k]; tmi[tid * K_TOP + k] = oi[k]; }
    }
    __syncthreads();
  }

  if (tid == 0) {
    const float M = red_m[0];
    const float Z = red_s[0];
    const float pos = __expf(sTgt - M) / Z;
    float num = 0.0f, den = 0.0f;
#pragma unroll
    for (int k = 0; k < K_TOP; ++k) {
      const float negp = __expf(tmv[k] - M) / Z;
      const float valid = (tmi[k] != tgt) ? 1.0f : 0.0f;
      num += valid * (pos - negp);
      den += valid;
    }
    row_out[row] = -(num / den) * mask[row];
  }
}

__global__ __launch_bounds__(BLOCK)
void cpo_finalize(const float* __restrict__ row_loss_masked,
                  const float* __restrict__ mask,
                  float* __restrict__ out, int n)
{
  __shared__ float sn[BLOCK], sd[BLOCK];
  float a = 0.0f, b = 0.0f;
  for (int i = threadIdx.x; i < n; i += BLOCK) { a += row_loss_masked[i]; b += mask[i]; }
  sn[threadIdx.x] = a; sd[threadIdx.x] = b;
  __syncthreads();
  for (int off = BLOCK / 2; off > 0; off >>= 1) {
    if (threadIdx.x < off) {
      sn[threadIdx.x] += sn[threadIdx.x + off];
      sd[threadIdx.x] += sd[threadIdx.x + off];
    }
    __syncthreads();
  }
  if (threadIdx.x == 0) out[0] = sn[0] / sd[0];
}

extern "C" void kernel_launch(void* const* d_in, const int* in_sizes, int n_in,
                              void* d_out, int out_size, void* d_ws, size_t ws_size,
                              hipStream_t stream) {
  const float* logits = (const float*)d_in[0];   // [B,S,V] fp32
  const int*   target = (const int*)d_in[1];     // [B,S]   int
  const float* mask   = (const float*)d_in[2];   // [B,S]   fp32
  float* out = (float*)d_out;                    // scalar
  float* row_loss = (float*)d_ws;                // [rows] scratch

  const int rows = in_sizes[1];                  // B*S = 4096
  const int V    = in_sizes[0] / rows;           // 50257

  cpo_row_kernel<<<rows, BLOCK, 0, stream>>>(logits, target, mask, row_loss, V);
  cpo_finalize<<<1, BLOCK, 0, stream>>>(row_loss, mask, out, rows);
}